// DGLGraphConv_37709812859403
// MI455X (gfx1250) — compile-verified
//
#include <hip/hip_runtime.h>

// ---------------------------------------------------------------------------
// GraphConv: h[dst] += feat[src] * w  (E=1.6M, D=64), then out = h @ W + b.
// h is accumulated in-place in d_out (each GEMM wave owns a disjoint 16-row
// strip and finishes all reads before its stores, so in-place is safe).
// ---------------------------------------------------------------------------

typedef float v2f __attribute__((ext_vector_type(2)));
typedef float v8f __attribute__((ext_vector_type(8)));

// ---- Kernel 0: zero the accumulator (d_out) -------------------------------
__global__ __launch_bounds__(256) void zero_f4(float4* __restrict__ p, int n4) {
    int i = blockIdx.x * blockDim.x + threadIdx.x;
    if (i < n4) p[i] = make_float4(0.f, 0.f, 0.f, 0.f);
}

// ---- Kernel 1: per-edge gather-multiply + scatter-add ---------------------
// One wave (32 lanes) per edge: lane l handles features 2l, 2l+1.
// feat row load is a coalesced 256B burst; scatter uses hw f32 atomics
// (global_atomic_add_f32) that land in the 192MB L2 (h is only 25.6MB).
__global__ __launch_bounds__(256) void edge_scatter(
    const float* __restrict__ feat,
    const float* __restrict__ edge_w,
    const int*   __restrict__ src,
    const int*   __restrict__ dst,
    float*       __restrict__ h,
    int n_edges)
{
    int gtid = blockIdx.x * blockDim.x + threadIdx.x;
    int e    = gtid >> 5;          // wave index == edge index
    int lane = gtid & 31;
    if (e >= n_edges) return;

    int   s = src[e];
    int   d = dst[e];
    float w = edge_w[e];

    const float2* frow = (const float2*)(feat + (size_t)s * 64);
    float2 f = frow[lane];

    float* base = h + (size_t)d * 64 + lane * 2;
    unsafeAtomicAdd(base + 0, f.x * w);
    unsafeAtomicAdd(base + 1, f.y * w);
}

// ---- Kernel 2: in-place GEMM  out[100000,64] = h @ W[64,64] + bias --------
// One wave per 16-row strip; 4 column tiles of 16; K=64 in steps of 4 using
// V_WMMA_F32_16X16X4_F32 (f32 in / f32 acc, matches reference numerics).
__global__ __launch_bounds__(256) void gemm_wmma_f32(
    float* __restrict__ h_inout,        // [n_rows, 64], overwritten with result
    const float* __restrict__ W,        // [64, 64] row-major
    const float* __restrict__ bias,     // [64]
    int n_rows)
{
    int gtid = blockIdx.x * blockDim.x + threadIdx.x;
    int wave = gtid >> 5;
    int lane = gtid & 31;
    int row0 = wave * 16;
    if (row0 + 16 > n_rows) return;

    int half = lane >> 4;               // 0: lanes 0-15, 1: lanes 16-31
    int lm   = lane & 15;

    v8f acc0 = {}, acc1 = {}, acc2 = {}, acc3 = {};

    // A: lane holds h[row0+lm][k0 + 2*half + {0,1}]
    const float* arow = h_inout + (size_t)(row0 + lm) * 64 + 2 * half;

    #pragma unroll
    for (int k0 = 0; k0 < 64; k0 += 4) {
        v2f a;
        a.x = arow[k0 + 0];
        a.y = arow[k0 + 1];

        // B: VGPR j holds W[k0 + 2*half + j][col] (rows j, j+2 pattern)
        const float* brow = W + (size_t)(k0 + 2 * half) * 64 + lm;
        v2f b0, b1, b2, b3;
        b0.x = brow[0];      b0.y = brow[64];        // cols  0..15
        b1.x = brow[16];     b1.y = brow[64 + 16];   // cols 16..31
        b2.x = brow[32];     b2.y = brow[64 + 32];   // cols 32..47
        b3.x = brow[48];     b3.y = brow[64 + 48];   // cols 48..63

        acc0 = __builtin_amdgcn_wmma_f32_16x16x4_f32(false, a, false, b0,
                                                     (short)0, acc0, false, false);
        acc1 = __builtin_amdgcn_wmma_f32_16x16x4_f32(false, a, false, b1,
                                                     (short)0, acc1, false, false);
        acc2 = __builtin_amdgcn_wmma_f32_16x16x4_f32(false, a, false, b2,
                                                     (short)0, acc2, false, false);
        acc3 = __builtin_amdgcn_wmma_f32_16x16x4_f32(false, a, false, b3,
                                                     (short)0, acc3, false, false);
    }

    // C/D: VGPR j -> row (row0 + j + 8*half), col (ct*16 + lm)
    float bv0 = bias[lm];
    float bv1 = bias[16 + lm];
    float bv2 = bias[32 + lm];
    float bv3 = bias[48 + lm];

    #pragma unroll
    for (int j = 0; j < 8; ++j) {
        float* orow = h_inout + (size_t)(row0 + j + 8 * half) * 64 + lm;
        orow[0]  = acc0[j] + bv0;
        orow[16] = acc1[j] + bv1;
        orow[32] = acc2[j] + bv2;
        orow[48] = acc3[j] + bv3;
    }
}

// ---------------------------------------------------------------------------
extern "C" void kernel_launch(void* const* d_in, const int* in_sizes, int n_in,
                              void* d_out, int out_size, void* d_ws, size_t ws_size,
                              hipStream_t stream) {
    const float* feat   = (const float*)d_in[0];
    const float* edge_w = (const float*)d_in[1];
    const int*   src    = (const int*)  d_in[2];
    const int*   dst    = (const int*)  d_in[3];
    const float* W      = (const float*)d_in[4];
    const float* bias   = (const float*)d_in[5];
    float*       out    = (float*)d_out;

    const int D       = 64;
    const int n_nodes = in_sizes[0] / D;     // 100000
    const int n_edges = in_sizes[1];         // 1600000
    const int total   = n_nodes * D;         // 6.4M floats

    // 1) zero accumulator (d_out doubles as h)
    {
        int n4 = total / 4;
        int blocks = (n4 + 255) / 256;
        zero_f4<<<blocks, 256, 0, stream>>>((float4*)out, n4);
    }

    // 2) edge gather-multiply-scatter: one wave per edge, 8 waves per block
    {
        int blocks = (n_edges + 7) / 8;
        edge_scatter<<<blocks, 256, 0, stream>>>(feat, edge_w, src, dst, out, n_edges);
    }

    // 3) in-place WMMA GEMM + bias: one wave per 16-row strip
    {
        int strips = n_nodes / 16;           // 6250 (exact)
        int blocks = (strips + 7) / 8;
        gemm_wmma_f32<<<blocks, 256, 0, stream>>>(out, W, bias, n_nodes);
    }
}